// ProbabilisticPrediction_48885317763374
// MI455X (gfx1250) — compile-verified
//
#include <hip/hip_runtime.h>
#include <hip/hip_bf16.h>
#include <math.h>

// Problem constants from the reference.
#define Bn   4
#define QLn  256
#define CLn  512
#define DIMn 128      // XD == ED == DIM == 128
#define YDn  16

typedef __attribute__((ext_vector_type(2))) float v2f;
typedef __attribute__((ext_vector_type(8))) float v8f;

// Async global->LDS path (gfx1250): use if the toolchain exposes the builtins.
#if defined(__HIP_DEVICE_COMPILE__) &&                                        \
    __has_builtin(__builtin_amdgcn_global_load_async_to_lds_b32) &&           \
    __has_builtin(__builtin_amdgcn_s_wait_asynccnt)
#define HAVE_ASYNC_LDS 1
#else
#define HAVE_ASYNC_LDS 0
#endif

__device__ __forceinline__ void stage_f32_to_lds(const float* src,
                                                 float* dst_lds) {
#if HAVE_ASYNC_LDS
  // Prototype (from hipcc diagnostic): param0 is addrspace(1) int* (global
  // src), param1 is addrspace(3) int* (LDS dst), then imm offset, imm cpol.
  __builtin_amdgcn_global_load_async_to_lds_b32(
      (__attribute__((address_space(1))) int*)src,
      (__attribute__((address_space(3))) int*)dst_lds,
      /*offset=*/0, /*cpol=*/0);
#else
  *dst_lds = *src;
#endif
}

__device__ __forceinline__ void wait_lds_staging() {
#if HAVE_ASYNC_LDS
  __builtin_amdgcn_s_wait_asynccnt(0);
#endif
  __syncthreads();
}

// ---------------------------------------------------------------------------
// deterministic hash helpers (stand-in for JAX RNG streams; compile-only env)
// ---------------------------------------------------------------------------
__device__ __forceinline__ unsigned hash_u32(unsigned x) {
  x *= 2654435761u; x ^= x >> 16; x *= 0x45d9f3bu; x ^= x >> 16;
  return x;
}
__device__ __forceinline__ float hash_unit(unsigned x) {  // (0,1)
  return (float)((hash_u32(x) >> 8) + 1u) * (1.0f / 16777217.0f);
}

// ---------------------------------------------------------------------------
// Kernel 1: P[M,128] = X[M,128] @ W[128,128]  via V_WMMA_F32_16X16X4_F32
// grid = (M/128, 8), block = 256 (8 waves; one 16x16 output tile per wave).
// The 128x16 W column-tile is staged TRANSPOSED in LDS (async-to-LDS when
// available) and shared by all 8 waves: each WMMA step then costs exactly
// one global_load_b64 (A) + one ds_load_b64 (B).
// ---------------------------------------------------------------------------
__global__ void gemm128_wmma_f32(const float* __restrict__ X,
                                 const float* __restrict__ W,
                                 float* __restrict__ P) {
  __shared__ float Wt[16][132];            // Wt[n][k], pad 132 to dodge bank conflicts
  const int tid   = threadIdx.x;
  const int lane  = tid & 31;              // wave32
  const int wave  = tid >> 5;              // 0..7
  const int mtile = blockIdx.x * 8 + wave;
  const int ntile = blockIdx.y;            // 0..7
  const int lo    = lane & 15;             // row (A) / col (B,C,D)
  const int kg    = lane >> 4;             // K-pair select

  // --- stage W[:, ntile*16 : ntile*16+16] transposed into LDS --------------
  for (int e = tid; e < 16 * DIMn; e += 256) {
    const int n = e & 15, k = e >> 4;      // consecutive tid -> consecutive n
    stage_f32_to_lds(&W[(size_t)k * DIMn + ntile * 16 + n], &Wt[n][k]);
  }
  wait_lds_staging();

  // --- 16x16 tile accumulate over K=128 (32 x K=4 WMMA issues) -------------
  v8f acc = {};
  const float* xrow = X + (size_t)(mtile * 16 + lo) * DIMn;
  __builtin_prefetch(xrow, 0, 1);          // -> global_prefetch_b8
#pragma unroll 4
  for (int ks = 0; ks < 32; ++ks) {
    const int k0 = ks * 4 + 2 * kg;        // even -> 8B aligned
    v2f a, b;
    const float2 av = *reinterpret_cast<const float2*>(xrow + k0);       // b64
    const float2 bv = *reinterpret_cast<const float2*>(&Wt[lo][k0]);     // ds b64
    a.x = av.x; a.y = av.y;
    b.x = bv.x; b.y = bv.y;
    acc = __builtin_amdgcn_wmma_f32_16x16x4_f32(
        /*neg_a=*/false, a, /*neg_b=*/false, b,
        /*c_mod=*/(short)0, acc, /*reuse_a=*/false, /*reuse_b=*/false);
  }

  // D layout: lanes 0-15 -> rows r, lanes 16-31 -> rows 8+r; col = lo
  const int rbase = mtile * 16 + kg * 8;
#pragma unroll
  for (int r = 0; r < 8; ++r)
    P[(size_t)(rbase + r) * DIMn + ntile * 16 + lo] = acc[r];
}

// ---------------------------------------------------------------------------
// Kernel 2: categorical sample per (b,q) row via Gumbel-max over CL logits.
// grid = B*QL, block = 32 (one wave per row)
// ---------------------------------------------------------------------------
__global__ void categorical_gumbel(const float* __restrict__ logits,
                                   int* __restrict__ idx) {
  const int row  = blockIdx.x;             // b*QL + q
  const int lane = threadIdx.x;
  const float* lrow = logits + (size_t)row * CLn;

  float best = -INFINITY; int bi = 0;
  for (int c = lane; c < CLn; c += 32) {
    const float u = hash_unit((unsigned)(row * CLn + c));
    const float g = -__logf(-__logf(u));
    const float v = lrow[c] + g;
    if (v > best) { best = v; bi = c; }
  }
#pragma unroll
  for (int off = 16; off > 0; off >>= 1) {
    const float ov = __shfl_xor(best, off, 32);
    const int   oi = __shfl_xor(bi,   off, 32);
    if (ov > best || (ov == best && oi < bi)) { best = ov; bi = oi; }
  }
  if (lane == 0) idx[row] = bi;
}

// ---------------------------------------------------------------------------
// Kernel 3: fused gather-first MDN tail.
// For 16 (b,q) rows per wave:
//   h1 = relu(qp[bq] + fp[b, idx[bq]] + b1)          (staged in LDS)
//   h2 = relu(h1 @ W2 + b2)                          (WMMA, staged in LDS)
//   out = h2 @ Wout + bout ; y = mu + softplus(s)*eps
// grid = (B*QL)/16, block = 32
// ---------------------------------------------------------------------------
__global__ void mdn_tail_wmma(const float* __restrict__ qp,
                              const float* __restrict__ fp,
                              const int*   __restrict__ idx,
                              const float* __restrict__ b1,
                              const float* __restrict__ W2,
                              const float* __restrict__ b2,
                              const float* __restrict__ Wout,
                              const float* __restrict__ bout,
                              float* __restrict__ y) {
  __shared__ float As[16][DIMn];   // h1 tile
  __shared__ float Hs[16][DIMn];   // h2 tile
  const int lane = threadIdx.x;
  const int lo = lane & 15, kg = lane >> 4;
  const int tileBase = blockIdx.x * 16;    // first bq of this tile

  // --- build h1 tile -------------------------------------------------------
  for (int t = lane; t < 16 * DIMn; t += 32) {
    const int row = t >> 7, col = t & (DIMn - 1);
    const int bq = tileBase + row;
    const int b  = bq >> 8;                // QL = 256
    const int c  = idx[bq];
    const float v = qp[(size_t)bq * DIMn + col] +
                    fp[((size_t)b * CLn + c) * DIMn + col] + b1[col];
    As[row][col] = v > 0.0f ? v : 0.0f;
  }
  __syncthreads();

  // --- h2 = relu(h1 @ W2 + b2) --------------------------------------------
#pragma unroll
  for (int ntile = 0; ntile < 8; ++ntile) {
    const int n = ntile * 16 + lo;
    v8f acc;
    const float bias = b2[n];
#pragma unroll
    for (int r = 0; r < 8; ++r) acc[r] = bias;   // bias broadcast over rows
#pragma unroll 4
    for (int ks = 0; ks < 32; ++ks) {
      const int k0 = ks * 4 + 2 * kg;
      v2f a, b;
      const float2 av = *reinterpret_cast<const float2*>(&As[lo][k0]);
      a.x = av.x; a.y = av.y;
      b.x = W2[(size_t)k0 * DIMn + n];
      b.y = W2[(size_t)(k0 + 1) * DIMn + n];
      acc = __builtin_amdgcn_wmma_f32_16x16x4_f32(
          false, a, false, b, (short)0, acc, false, false);
    }
#pragma unroll
    for (int r = 0; r < 8; ++r) {
      const float v = acc[r];
      Hs[kg * 8 + r][n] = v > 0.0f ? v : 0.0f;
    }
  }
  __syncthreads();

  // --- out = h2 @ Wout + bout ; mu in cols 0..15, s in cols 16..31 --------
  v8f accMu, accS;
  const float bmu0 = bout[lo], bs0 = bout[16 + lo];
#pragma unroll
  for (int r = 0; r < 8; ++r) { accMu[r] = bmu0; accS[r] = bs0; }
#pragma unroll 4
  for (int ks = 0; ks < 32; ++ks) {
    const int k0 = ks * 4 + 2 * kg;
    v2f a, bm, bs;
    const float2 av = *reinterpret_cast<const float2*>(&Hs[lo][k0]);
    a.x = av.x; a.y = av.y;
    bm.x = Wout[(size_t)k0 * (2 * YDn) + lo];
    bm.y = Wout[(size_t)(k0 + 1) * (2 * YDn) + lo];
    bs.x = Wout[(size_t)k0 * (2 * YDn) + 16 + lo];
    bs.y = Wout[(size_t)(k0 + 1) * (2 * YDn) + 16 + lo];
    accMu = __builtin_amdgcn_wmma_f32_16x16x4_f32(
        false, a, false, bm, (short)0, accMu, false, false);
    accS = __builtin_amdgcn_wmma_f32_16x16x4_f32(
        false, a, false, bs, (short)0, accS, false, false);
  }

  // --- sample: y = mu + softplus(max(-15,s)) * eps -------------------------
#pragma unroll
  for (int r = 0; r < 8; ++r) {
    const int bq = tileBase + kg * 8 + r;
    const float s = fmaxf(-15.0f, accS[r]);
    const float sig = (s > 20.0f) ? s : log1pf(__expf(s));
    const unsigned seed = (unsigned)(bq * YDn + lo);
    const float u1 = hash_unit(seed);
    const float u2 = hash_unit(seed ^ 0x9E3779B9u);
    const float eps = sqrtf(-2.0f * __logf(u1)) * __cosf(6.28318530718f * u2);
    y[(size_t)bq * YDn + lo] = accMu[r] + sig * eps;
  }
}

// ---------------------------------------------------------------------------
extern "C" void kernel_launch(void* const* d_in, const int* in_sizes, int n_in,
                              void* d_out, int out_size, void* d_ws, size_t ws_size,
                              hipStream_t stream) {
  const float* q      = (const float*)d_in[0];   // [B,QL,128]
  const float* f_emb  = (const float*)d_in[1];   // [B,CL,128]
  const float* att    = (const float*)d_in[2];   // [B,QL,CL]
  const float* W1q    = (const float*)d_in[3];   // [128,128]
  const float* W1f    = (const float*)d_in[4];   // [128,128]
  const float* b1     = (const float*)d_in[5];   // [128]
  const float* W2     = (const float*)d_in[6];   // [128,128]
  const float* b2     = (const float*)d_in[7];   // [128]
  const float* Wout   = (const float*)d_in[8];   // [128,32]
  const float* bout   = (const float*)d_in[9];   // [32]
  float* y = (float*)d_out;                      // [B,QL,16]

  // workspace layout
  const int Mq = Bn * QLn;                       // 1024
  const int Mf = Bn * CLn;                       // 2048
  float* qp  = (float*)d_ws;                     // [Mq,128]  512 KB
  float* fpb = qp + (size_t)Mq * DIMn;           // [Mf,128]  1 MB
  int*   idx = (int*)(fpb + (size_t)Mf * DIMn);  // [Mq]      4 KB

  gemm128_wmma_f32<<<dim3(Mq / 128, DIMn / 16), 256, 0, stream>>>(q, W1q, qp);
  gemm128_wmma_f32<<<dim3(Mf / 128, DIMn / 16), 256, 0, stream>>>(f_emb, W1f, fpb);
  categorical_gumbel<<<Mq, 32, 0, stream>>>(att, idx);
  mdn_tail_wmma<<<Mq / 16, 32, 0, stream>>>(qp, fpb, idx, b1, W2, b2, Wout, bout, y);
}